// FastText_40510131536002
// MI455X (gfx1250) — compile-verified
//
#include <hip/hip_runtime.h>
#include <math.h>

typedef __attribute__((ext_vector_type(2))) float v2f;
typedef __attribute__((ext_vector_type(8))) float v8f;

#define EDIM   512
#define BSZ    1024
#define SEQ    50
#define LEAVES 1024
#define DEPTH  10
#define NNODES 1023
#define NCOLS  2046   /* NNODES * 2 */
#define LDLOG  2048   /* padded logits stride */

#define BM 64
#define BN 64
#define BK 16
#define LDA_S 18      /* even -> 8B aligned float2 frags, conflict-free */
#define LDB_S 80      /* rows k and k+2 hit disjoint bank halves */

// ---------------------------------------------------------------------------
// CDNA5 async copy helpers (gfx1250): global -> LDS DMA, tracked by ASYNCcnt.
// LDS operand is the workgroup-relative byte offset (= low 32 bits of the
// generic pointer to a __shared__ object).
// ---------------------------------------------------------------------------
__device__ __forceinline__ uint32_t lds_off(const void* p) {
  return (uint32_t)(uintptr_t)p;
}
__device__ __forceinline__ void async_ld_b64(uint32_t lds, const float* g) {
  asm volatile("global_load_async_to_lds_b64 %0, %1, off"
               :: "v"(lds), "v"(g) : "memory");
}
__device__ __forceinline__ void async_ld_b128(uint32_t lds, const float* g) {
  asm volatile("global_load_async_to_lds_b128 %0, %1, off"
               :: "v"(lds), "v"(g) : "memory");
}
template <int N>
__device__ __forceinline__ void wait_async() {
  asm volatile("s_wait_asynccnt %0" :: "i"(N) : "memory");
}

// ---------------------------------------------------------------------------
// Kernel 1: embedding mean over SEQ tokens + LayerNorm over EDIM
// ---------------------------------------------------------------------------
__global__ __launch_bounds__(256) void k_embed_ln(const int* __restrict__ ids,
                                                  const float* __restrict__ emb,
                                                  float* __restrict__ xout) {
  const int b = blockIdx.x;
  const int tid = threadIdx.x;
  const int* idr = ids + b * SEQ;
  const int e0 = tid, e1 = tid + 256;
  float s0 = 0.f, s1 = 0.f;
  for (int s = 0; s < SEQ; ++s) {
    const float* row = emb + (size_t)idr[s] * EDIM;
    s0 += row[e0];
    s1 += row[e1];
  }
  s0 *= (1.0f / SEQ);
  s1 *= (1.0f / SEQ);

  __shared__ float rs[256];
  __shared__ float rq[256];
  rs[tid] = s0 + s1;
  rq[tid] = s0 * s0 + s1 * s1;
  __syncthreads();
  for (int st = 128; st > 0; st >>= 1) {
    if (tid < st) { rs[tid] += rs[tid + st]; rq[tid] += rq[tid + st]; }
    __syncthreads();
  }
  const float mean = rs[0] * (1.0f / EDIM);
  const float var  = rq[0] * (1.0f / EDIM) - mean * mean;
  const float rstd = rsqrtf(var + 1e-5f);
  float* xo = xout + (size_t)b * EDIM;
  xo[e0] = (s0 - mean) * rstd;
  xo[e1] = (s1 - mean) * rstd;
}

// ---------------------------------------------------------------------------
// Kernels 2/3: fp32 WMMA GEMM with async-LDS double buffering.
//   MODE 0: B = W_h (E x E row-major),            N = 512,  ldc = 512
//   MODE 1: B = theta_W (n,e,d) seen as (E x 2046), N = 2046, ldc = 2048
// Per wave per stage: A = 2x async b64, B = 1x b128 (MODE 0) / 2x b64 (MODE 1)
// ---------------------------------------------------------------------------
template <int MODE>
__global__ __launch_bounds__(256) void k_gemm(const float* __restrict__ A,
                                              const float* __restrict__ Bsrc,
                                              const float* __restrict__ bias,
                                              float* __restrict__ C) {
  __shared__ float As[2][BM * LDA_S];
  __shared__ float Bs[2][BK * LDB_S];

  const int tid  = threadIdx.x;
  const int lane = tid & 31;
  const int wave = tid >> 5;
  const int wm = wave & 3;   // 4 row sub-tiles of 16
  const int wn = wave >> 2;  // 2 col sub-tiles of 32
  const int rowBase = blockIdx.y * BM;
  const int colBase = blockIdx.x * BN;
  const int ldc   = (MODE == 0) ? EDIM : LDLOG;
  const int nreal = (MODE == 0) ? EDIM : NCOLS;
  constexpr int PREF = (MODE == 0) ? 3 : 4;   // async instrs per wave per stage

  // per-thread staging coordinates (stage-invariant)
  const int ar0 = tid >> 3,          ac0 = (tid & 7) << 1;          // A pair 0
  const int ar1 = (tid + 256) >> 3,  ac1 = ((tid + 256) & 7) << 1;  // A pair 1
  const int bk0 = tid >> 4,          bn0 = (tid & 15) << 2;         // B quad (MODE 0)
  const int pk0 = tid >> 5,          pn0 = (tid & 31) << 1;         // B pair 0 (MODE 1)
  const int pk1 = (tid + 256) >> 5,  pn1 = ((tid + 256) & 31) << 1; // B pair 1 (MODE 1)

  auto stage = [&](int buf, int k0) {
    async_ld_b64(lds_off(&As[buf][ar0 * LDA_S + ac0]),
                 A + (size_t)(rowBase + ar0) * EDIM + k0 + ac0);
    async_ld_b64(lds_off(&As[buf][ar1 * LDA_S + ac1]),
                 A + (size_t)(rowBase + ar1) * EDIM + k0 + ac1);
    if (MODE == 0) {
      async_ld_b128(lds_off(&Bs[buf][bk0 * LDB_S + bn0]),
                    Bsrc + (size_t)(k0 + bk0) * EDIM + colBase + bn0);
    } else {
      // column pair j = colBase + pn (node = j>>1, d = 0,1 contiguous);
      // clamp the single padded pair so every async issues with full EXEC
      int j0 = colBase + pn0; j0 = (j0 < NCOLS) ? j0 : (NCOLS - 2);
      int j1 = colBase + pn1; j1 = (j1 < NCOLS) ? j1 : (NCOLS - 2);
      async_ld_b64(lds_off(&Bs[buf][pk0 * LDB_S + pn0]),
                   Bsrc + (size_t)(j0 >> 1) * (2 * EDIM) + (size_t)(k0 + pk0) * 2);
      async_ld_b64(lds_off(&Bs[buf][pk1 * LDB_S + pn1]),
                   Bsrc + (size_t)(j1 >> 1) * (2 * EDIM) + (size_t)(k0 + pk1) * 2);
    }
  };

  v8f acc0 = {};
  v8f acc1 = {};

  const int mrow = wm * 16 + (lane & 15);
  const int klo  = (lane < 16) ? 0 : 2;   // K split across half-waves
  const int ncol = lane & 15;

  constexpr int NSTAGE = EDIM / BK;       // 32
  stage(0, 0);
  for (int it = 0; it < NSTAGE; ++it) {
    const int cur = it & 1;
    if (it + 1 < NSTAGE) {
      stage(cur ^ 1, (it + 1) * BK);      // prefetch next tile
      wait_async<PREF>();                 // current tile resident
    } else {
      wait_async<0>();
    }
    __syncthreads();

    const float* as = As[cur];
    const float* bs = Bs[cur];
    #pragma unroll
    for (int kk = 0; kk < BK; kk += 4) {
      v2f a = *(const v2f*)&as[mrow * LDA_S + kk + klo];
      const int bro = (kk + klo) * LDB_S;
      v2f b0, b1;
      b0.x = bs[bro + wn * 32 + ncol];
      b0.y = bs[bro + LDB_S + wn * 32 + ncol];
      b1.x = bs[bro + wn * 32 + 16 + ncol];
      b1.y = bs[bro + LDB_S + wn * 32 + 16 + ncol];
      acc0 = __builtin_amdgcn_wmma_f32_16x16x4_f32(false, a, false, b0,
                                                   (short)0, acc0, false, false);
      acc1 = __builtin_amdgcn_wmma_f32_16x16x4_f32(false, a, false, b1,
                                                   (short)0, acc1, false, false);
    }
    __syncthreads();   // everyone done reading before this buffer is re-staged
  }

  // epilogue: C/D layout VGPR v -> M=v (lanes 0-15) / v+8 (lanes 16-31)
  const int mbase = rowBase + wm * 16 + ((lane < 16) ? 0 : 8);
  #pragma unroll
  for (int v = 0; v < 8; ++v) {
    const int row = mbase + v;
    const int c0 = colBase + wn * 32 + ncol;
    const int c1 = c0 + 16;
    if (c0 < nreal) C[(size_t)row * ldc + c0] = acc0[v] + bias[c0];
    if (c1 < nreal) C[(size_t)row * ldc + c1] = acc1[v] + bias[c1];
  }
}

// ---------------------------------------------------------------------------
// Kernel 4: pairwise softmax + hierarchical path product
//   out[b, leaf] = prod_t sigmoid(l_bit - l_other) at node paths[leaf,t]
// ---------------------------------------------------------------------------
__global__ __launch_bounds__(256) void k_leafprod(const float* __restrict__ logits,
                                                  const int* __restrict__ paths,
                                                  const int* __restrict__ bits,
                                                  float* __restrict__ out) {
  const int b = blockIdx.x;
  const int tid = threadIdx.x;
  __shared__ float lrow[NCOLS];
  for (int j = tid; j < NCOLS; j += 256) lrow[j] = logits[(size_t)b * LDLOG + j];
  __syncthreads();

  for (int leaf = tid; leaf < LEAVES; leaf += 256) {
    float prod = 1.0f;
    #pragma unroll
    for (int t = 0; t < DEPTH; ++t) {
      const int node = paths[leaf * DEPTH + t];
      const int bit  = bits[leaf * DEPTH + t];
      const float l0 = lrow[2 * node];
      const float l1 = lrow[2 * node + 1];
      const float z = bit ? (l1 - l0) : (l0 - l1);
      prod *= 1.0f / (1.0f + __expf(-z));
    }
    out[(size_t)b * LEAVES + leaf] = prod;
  }
}

// ---------------------------------------------------------------------------
extern "C" void kernel_launch(void* const* d_in, const int* in_sizes, int n_in,
                              void* d_out, int out_size, void* d_ws, size_t ws_size,
                              hipStream_t stream) {
  (void)in_sizes; (void)n_in; (void)out_size; (void)ws_size;

  const int*   ids   = (const int*)d_in[0];
  const float* emb   = (const float*)d_in[1];
  const float* W_h   = (const float*)d_in[2];
  const float* b_h   = (const float*)d_in[3];
  const float* thW   = (const float*)d_in[4];
  const float* thb   = (const float*)d_in[5];
  const int*   paths = (const int*)d_in[6];
  const int*   bits  = (const int*)d_in[7];
  float* out = (float*)d_out;

  float* buf0   = (float*)d_ws;                       // 1024*512
  float* buf1   = buf0 + (size_t)BSZ * EDIM;          // 1024*512
  float* logits = buf1 + (size_t)BSZ * EDIM;          // 1024*2048

  // 1) embedding mean + layernorm
  k_embed_ln<<<BSZ, 256, 0, stream>>>(ids, emb, buf0);

  // 2) 10x hidden GEMM chain (fp32 WMMA), ping-pong
  float* pin = buf0;
  float* pout = buf1;
  for (int i = 0; i < 10; ++i) {
    dim3 g(EDIM / BN, BSZ / BM);   // (8, 16)
    k_gemm<0><<<g, 256, 0, stream>>>(pin, W_h, b_h, pout);
    float* t = pin; pin = pout; pout = t;
  }

  // 3) projection onto tree nodes: logits (1024 x 2046, stride 2048)
  {
    dim3 g(LDLOG / BN, BSZ / BM);  // (32, 16)
    k_gemm<1><<<g, 256, 0, stream>>>(pin, thW, thb, logits);
  }

  // 4) softmax-pair + path product
  k_leafprod<<<BSZ, 256, 0, stream>>>(logits, paths, bits, out);
}